// MyModel_71975061946506
// MI455X (gfx1250) — compile-verified
//
#include <hip/hip_runtime.h>

// out = x + 4095.0f * y   (reference: x + (N-1)*y, N = 4096 static)
//
// HBM-bandwidth-bound streaming FMA (AI ~ 0.17 FLOP/B -> floor ~17us at
// 23.3 TB/s). MI455X tuning: 128-bit VMEM ops, CDNA5 non-temporal cache
// hints (384 MiB single-touch stream > 192 MB L2), 8 independent b128 loads
// in flight per lane, wave32-friendly 256-thread blocks, and a fully
// straight-line hot kernel (no predication; exact-division decomposition,
// remainder handled by a separate tail kernel that never launches for this
// shape).

typedef __attribute__((ext_vector_type(4))) float v4f;

#define TPB   256u   // 8 wave32 waves per block
#define VPT   4u     // float4 tiles per thread (64 B/thread)

__global__ __launch_bounds__(TPB) void fma_stream_kernel(
    const v4f* __restrict__ x,
    const v4f* __restrict__ y,
    v4f* __restrict__ out,
    float scale)
{
    const unsigned base = blockIdx.x * (TPB * VPT) + threadIdx.x;

    v4f xv[VPT];
    v4f yv[VPT];

    // 8 outstanding NT b128 loads per lane; no guards -> clause-able.
#pragma unroll
    for (unsigned k = 0; k < VPT; ++k)
        xv[k] = __builtin_nontemporal_load(&x[base + k * TPB]);
#pragma unroll
    for (unsigned k = 0; k < VPT; ++k)
        yv[k] = __builtin_nontemporal_load(&y[base + k * TPB]);

#pragma unroll
    for (unsigned k = 0; k < VPT; ++k) {
        v4f r;
        r.x = fmaf(scale, yv[k].x, xv[k].x);
        r.y = fmaf(scale, yv[k].y, xv[k].y);
        r.z = fmaf(scale, yv[k].z, xv[k].z);
        r.w = fmaf(scale, yv[k].w, xv[k].w);
        __builtin_nontemporal_store(r, &out[base + k * TPB]);
    }
}

// Scalar tail for any elements not covered by full blocks (never runs for
// the reference shape 4096x8192, which divides exactly).
__global__ __launch_bounds__(TPB) void fma_tail_kernel(
    const float* __restrict__ x,
    const float* __restrict__ y,
    float* __restrict__ out,
    float scale,
    unsigned start,
    unsigned n)
{
    const unsigned i = start + blockIdx.x * TPB + threadIdx.x;
    if (i < n)
        out[i] = fmaf(scale, y[i], x[i]);
}

extern "C" void kernel_launch(void* const* d_in, const int* in_sizes, int n_in,
                              void* d_out, int out_size, void* d_ws, size_t ws_size,
                              hipStream_t stream) {
    (void)n_in; (void)d_ws; (void)ws_size; (void)out_size;

    const float* x   = (const float*)d_in[0];
    const float* y   = (const float*)d_in[1];
    float*       out = (float*)d_out;

    const unsigned n_elems = (unsigned)in_sizes[0];   // 4096 * 8192
    const unsigned nvec    = n_elems / 4u;            // float4 count

    // Static leading dim from the reference: N = 4096 -> scale = N - 1.
    const float scale = 4095.0f;

    const unsigned per_block   = TPB * VPT;            // 1024 float4 per block
    const unsigned full_blocks = nvec / per_block;     // 8192 for this shape

    if (full_blocks > 0) {
        fma_stream_kernel<<<full_blocks, TPB, 0, stream>>>(
            (const v4f*)x, (const v4f*)y, (v4f*)out, scale);
    }

    const unsigned covered = full_blocks * per_block * 4u;  // elements done
    if (covered < n_elems) {
        const unsigned rem = n_elems - covered;
        fma_tail_kernel<<<(rem + TPB - 1) / TPB, TPB, 0, stream>>>(
            x, y, out, scale, covered, n_elems);
    }
}